// RetNext_40346922779120
// MI455X (gfx1250) — compile-verified
//
#include <hip/hip_runtime.h>

typedef __attribute__((ext_vector_type(2))) float v2f;
typedef __attribute__((ext_vector_type(8))) float v8f;

#define BM 64
#define BN 64
#define BK 32

#define A_STRIDE    36   // MK layout row stride in floats (144B, 16B-aligned)
#define B_NK_STRIDE 36   // NK layout (transB) row stride
#define B_KN_STRIDE 68   // KN layout (!transB) row stride (272B, 16B-aligned)
#define BUF_ELEMS   2304 // max(64*36, 32*68)

// One 16-byte global->LDS async copy (no VGPR staging, tracked by ASYNCcnt).
__device__ __forceinline__ void async_copy16(unsigned lds_off, const float* gptr) {
    asm volatile("global_load_async_to_lds_b128 %0, %1, off"
                 :: "v"(lds_off), "v"(gptr)
                 : "memory");
}

// Generic fp32 WMMA GEMM:
//   C[m][n] = sum_k A[m*lda+k] * (TRANSB ? B[n*ldb+k] : B[k*ldb+n]) (+ bias[n])
// TRANSC: store C[n*ldc+m]. Batched over blockIdx.z, z = b*nh + h.
template<int TRANSB, int TRANSC, int HASBIAS>
__global__ __launch_bounds__(128)
void retnext_gemm_f32_wmma(const float* __restrict__ A,
                           const float* __restrict__ B,
                           const float* __restrict__ bias,
                           float* __restrict__ C,
                           int M, int N, int K,
                           int lda, int ldb, int ldc,
                           long long sAb, long long sAh,
                           long long sBb, long long sBh,
                           long long sCb, long long sCh,
                           int nh)
{
    __shared__ __align__(16) float As[2][BUF_ELEMS];  // [m][k], stride A_STRIDE
    __shared__ __align__(16) float Bs[2][BUF_ELEMS];  // NK or KN layout

    const int z  = blockIdx.z;
    const int bb = z / nh;
    const int hh = z % nh;
    A += bb * sAb + hh * sAh;
    B += bb * sBb + hh * sBh;
    C += bb * sCb + hh * sCh;

    const int nBase = blockIdx.x * BN;
    const int mBase = blockIdx.y * BM;

    const int tid   = threadIdx.x;
    const int lane  = tid & 31;
    const int wave  = tid >> 5;          // 0..3
    const int lmod  = lane & 15;
    const int lhalf = lane >> 4;         // 0 or 1
    const int wm    = (wave >> 1) * 32;  // wave's row offset in 64x64 tile
    const int wn    = (wave & 1) * 32;   // wave's col offset

    const unsigned asBase = (unsigned)(size_t)&As[0][0];  // LDS byte offset
    const unsigned bsBase = (unsigned)(size_t)&Bs[0][0];

    // Issue 8 async B128 copies (4 for A tile, 4 for B tile) into buffer `buf`.
    auto stage_async = [&](int k0, int buf) {
#pragma unroll
        for (int p = 0; p < 4; ++p) {
            int f   = tid + p * 128;         // 512 x float4 = 64 rows * 8 groups
            int row = f >> 3;                // m
            int cg  = (f & 7) * 4;           // k
            const float* g = A + (size_t)(mBase + row) * lda + k0 + cg;
            unsigned l = asBase + (unsigned)((buf * BUF_ELEMS + row * A_STRIDE + cg) * 4);
            async_copy16(l, g);
        }
        if (TRANSB) {
            // source B[n][k] -> Bs_nk[n][k]: pure row copy
#pragma unroll
            for (int p = 0; p < 4; ++p) {
                int f   = tid + p * 128;
                int row = f >> 3;            // n
                int cg  = (f & 7) * 4;       // k
                const float* g = B + (size_t)(nBase + row) * ldb + k0 + cg;
                unsigned l = bsBase + (unsigned)((buf * BUF_ELEMS + row * B_NK_STRIDE + cg) * 4);
                async_copy16(l, g);
            }
        } else {
            // source B[k][n] -> Bs_kn[k][n]: pure row copy
#pragma unroll
            for (int p = 0; p < 4; ++p) {
                int f   = tid + p * 128;     // 512 x float4 = 32 rows * 16 groups
                int row = f >> 4;            // k
                int cg  = (f & 15) * 4;      // n
                const float* g = B + (size_t)(k0 + row) * ldb + nBase + cg;
                unsigned l = bsBase + (unsigned)((buf * BUF_ELEMS + row * B_KN_STRIDE + cg) * 4);
                async_copy16(l, g);
            }
        }
    };

    v8f acc[4];
#pragma unroll
    for (int i = 0; i < 4; ++i) acc[i] = (v8f)0.0f;

    const int nChunks = K / BK;
    stage_async(0, 0);

    for (int c = 0; c < nChunks; ++c) {
        const int cur = c & 1;

        if (c + 1 < nChunks) {
            stage_async((c + 1) * BK, cur ^ 1);
            // chunk c's 8 asyncs complete (in order); chunk c+1's 8 stay in flight
            asm volatile("s_wait_asynccnt 0x8" ::: "memory");
        } else {
            asm volatile("s_wait_asynccnt 0x0" ::: "memory");
        }
        __syncthreads();

        const float* Ab = &As[cur][0];
        const float* Bb = &Bs[cur][0];

#pragma unroll
        for (int kk = 0; kk < BK; kk += 4) {
            const int ka = kk + lhalf * 2;
            v2f a0, a1, b0, b1;
            a0[0] = Ab[(wm + lmod) * A_STRIDE + ka];
            a0[1] = Ab[(wm + lmod) * A_STRIDE + ka + 1];
            a1[0] = Ab[(wm + 16 + lmod) * A_STRIDE + ka];
            a1[1] = Ab[(wm + 16 + lmod) * A_STRIDE + ka + 1];
            if (TRANSB) {
                b0[0] = Bb[(wn + lmod) * B_NK_STRIDE + ka];
                b0[1] = Bb[(wn + lmod) * B_NK_STRIDE + ka + 1];
                b1[0] = Bb[(wn + 16 + lmod) * B_NK_STRIDE + ka];
                b1[1] = Bb[(wn + 16 + lmod) * B_NK_STRIDE + ka + 1];
            } else {
                b0[0] = Bb[ka * B_KN_STRIDE + wn + lmod];
                b0[1] = Bb[(ka + 1) * B_KN_STRIDE + wn + lmod];
                b1[0] = Bb[ka * B_KN_STRIDE + wn + 16 + lmod];
                b1[1] = Bb[(ka + 1) * B_KN_STRIDE + wn + 16 + lmod];
            }
            acc[0] = __builtin_amdgcn_wmma_f32_16x16x4_f32(false, a0, false, b0, (short)0, acc[0], false, false);
            acc[1] = __builtin_amdgcn_wmma_f32_16x16x4_f32(false, a0, false, b1, (short)0, acc[1], false, false);
            acc[2] = __builtin_amdgcn_wmma_f32_16x16x4_f32(false, a1, false, b0, (short)0, acc[2], false, false);
            acc[3] = __builtin_amdgcn_wmma_f32_16x16x4_f32(false, a1, false, b1, (short)0, acc[3], false, false);
        }
        __syncthreads();
    }

    // ---- epilogue: bias + store (normal or transposed) ----
#pragma unroll
    for (int i = 0; i < 2; ++i) {
#pragma unroll
        for (int j = 0; j < 2; ++j) {
            v8f r = acc[i * 2 + j];
            const int n = nBase + wn + j * 16 + lmod;
            float bv = 0.0f;
            if (HASBIAS) bv = bias[n];
            const int mB = mBase + wm + i * 16 + lhalf * 8;
            if (!TRANSC) {
#pragma unroll
                for (int e = 0; e < 8; ++e)
                    C[(size_t)(mB + e) * ldc + n] = r[e] + bv;
            } else {
#pragma unroll
                for (int e = 0; e < 8; ++e)
                    C[(size_t)n * ldc + mB + e] = r[e] + bv;
            }
        }
    }
}

extern "C" void kernel_launch(void* const* d_in, const int* in_sizes, int n_in,
                              void* d_out, int out_size, void* d_ws, size_t ws_size,
                              hipStream_t stream)
{
    const float* x  = (const float*)d_in[0];
    const float* Wk = (const float*)d_in[1];
    const float* bk = (const float*)d_in[2];
    const float* Wq = (const float*)d_in[3];
    const float* bq = (const float*)d_in[4];
    const float* Wv = (const float*)d_in[5];
    const float* bv = (const float*)d_in[6];
    float* out = (float*)d_out;

    const int Bsz = 4, T = 4096, D = 2048, H = 4, Dh = 512;
    const int Mtot = Bsz * T;  // 16384

    // workspace (fp32): KT (D x Mtot), Q (Mtot x D), V (Mtot x D), S (B*H x Dh x Dh)
    float* KT = (float*)d_ws;
    float* Q  = KT + (size_t)D * Mtot;
    float* V  = Q  + (size_t)Mtot * D;
    float* S  = V  + (size_t)Mtot * D;

    dim3 blk(128);

    // Phase 1a: KT = (x @ Wk^T + bk)^T   -> KT[d][m], ldc = Mtot
    retnext_gemm_f32_wmma<1, 1, 1><<<dim3(D / BN, Mtot / BM, 1), blk, 0, stream>>>(
        x, Wk, bk, KT, Mtot, D, D, D, D, Mtot,
        0, 0, 0, 0, 0, 0, 1);

    // Phase 1b: Q = x @ Wq^T + bq
    retnext_gemm_f32_wmma<1, 0, 1><<<dim3(D / BN, Mtot / BM, 1), blk, 0, stream>>>(
        x, Wq, bq, Q, Mtot, D, D, D, D, D,
        0, 0, 0, 0, 0, 0, 1);

    // Phase 1c: V = x @ Wv^T + bv
    retnext_gemm_f32_wmma<1, 0, 1><<<dim3(D / BN, Mtot / BM, 1), blk, 0, stream>>>(
        x, Wv, bv, V, Mtot, D, D, D, D, D,
        0, 0, 0, 0, 0, 0, 1);

    // Phase 2: states[b,h] = KT[h*Dh:, b*T:] (Dh x T) @ V[b*T:, h*Dh:] (T x Dh)
    retnext_gemm_f32_wmma<0, 0, 0><<<dim3(Dh / BN, Dh / BM, Bsz * H), blk, 0, stream>>>(
        KT, V, nullptr, S, Dh, Dh, T,
        /*lda*/Mtot, /*ldb*/D, /*ldc*/Dh,
        /*sAb*/(long long)T,           /*sAh*/(long long)Dh * Mtot,
        /*sBb*/(long long)T * D,       /*sBh*/(long long)Dh,
        /*sCb*/(long long)H * Dh * Dh, /*sCh*/(long long)Dh * Dh,
        H);

    // Phase 3: out[b,h] = Q[b*T:, h*Dh:] (T x Dh) @ states[b,h] (Dh x Dh)
    retnext_gemm_f32_wmma<0, 0, 0><<<dim3(Dh / BN, T / BM, Bsz * H), blk, 0, stream>>>(
        Q, S, nullptr, out, T, Dh, Dh,
        /*lda*/D, /*ldb*/Dh, /*ldc*/Dh,
        /*sAb*/(long long)T * D,       /*sAh*/(long long)Dh,
        /*sBb*/(long long)H * Dh * Dh, /*sBh*/(long long)Dh * Dh,
        /*sCb*/(long long)H * T * Dh,  /*sCh*/(long long)T * Dh,
        H);
}